// QuantConv_18287970747074
// MI455X (gfx1250) — compile-verified
//
#include <hip/hip_runtime.h>
#include <hip/hip_bf16.h>

typedef __attribute__((ext_vector_type(2))) float v2f;
typedef __attribute__((ext_vector_type(8))) float v8f;

#define PI_HALF 1.57079632679489661923f

// ---------------------------------------------------------------------------
// Kernel 1: build the 16x16 REAL entangler unitary U from the 2x4 weights.
// Thread j evolves basis column e_j through 2 layers of (RY x4, CNOT ring).
// U stored row-major: U[row*16 + col].
// ---------------------------------------------------------------------------
__global__ void build_unitary(const float* __restrict__ wts, float* __restrict__ U) {
  int j = threadIdx.x;
  if (j >= 16) return;
  float v[16];
#pragma unroll
  for (int b = 0; b < 16; ++b) v[b] = (b == j) ? 1.0f : 0.0f;

#pragma unroll
  for (int l = 0; l < 2; ++l) {
    // RY(theta_q) on each qubit q (qubit q owns bit (3-q) of the index)
#pragma unroll
    for (int q = 0; q < 4; ++q) {
      float th = 0.5f * wts[l * 4 + q];
      float s, c;
      __sincosf(th, &s, &c);
      const int mask = 1 << (3 - q);
#pragma unroll
      for (int b = 0; b < 16; ++b) {
        if (!(b & mask)) {
          float a0 = v[b], a1 = v[b | mask];
          v[b]        = c * a0 - s * a1;   // new0 = c*old0 - s*old1
          v[b | mask] = s * a0 + c * a1;   // new1 = s*old0 + c*old1
        }
      }
    }
    // CNOT ring: CNOT(0,1), CNOT(1,2), CNOT(2,3), CNOT(3,0)
#pragma unroll
    for (int q = 0; q < 4; ++q) {
      const int cm = 1 << (3 - q);
      const int tm = 1 << (3 - ((q + 1) & 3));
      float nv[16];
#pragma unroll
      for (int b = 0; b < 16; ++b) nv[b] = v[(b & cm) ? (b ^ tm) : b];
#pragma unroll
      for (int b = 0; b < 16; ++b) v[b] = nv[b];
    }
  }
#pragma unroll
  for (int b = 0; b < 16; ++b) U[b * 16 + j] = v[b];
}

// ---------------------------------------------------------------------------
// Kernel 2: one wave32 processes a tile of 16 windows via 12 f32 WMMAs.
//   GEMM1: OutRe^T = Ae(16x16, parity-masked U) x W^T ; OutIm^T = Ao x W^T
//   probs = re^2 + im^2 (C/D layout: VGPR=component half, lane=window)
//   GEMM2: EZ^T = Az(Z-signs) x probs^T  (probs rearranged by half-shuffles)
// ---------------------------------------------------------------------------
__global__ __launch_bounds__(256) void quanv_kernel(
    const float* __restrict__ x, const float* __restrict__ U,
    float* __restrict__ out, int tiles_total) {
  const int lane = threadIdx.x & 31;
  const int wave = threadIdx.x >> 5;
  const int m  = lane & 15;   // A: M index (n_out / q) ; B: window index
  const int hi = lane >> 4;   // which K-half this lane supplies

  // ---- constant A operands (per-lane, loaded once) ----
  v2f Ae[4], Ao[4], Az[4];
#pragma unroll
  for (int kk = 0; kk < 4; ++kk) {
#pragma unroll
    for (int v = 0; v < 2; ++v) {
      const int b = 4 * kk + v + 2 * hi;       // K index (state component)
      const float u = U[m * 16 + b];
      const int par = __popc(b) & 1;
      Ae[kk][v] = par ? 0.0f : u;              // even-popcount columns of U
      Ao[kk][v] = par ? u : 0.0f;              // odd-popcount columns of U
      Az[kk][v] = (m < 4) ? (1.0f - 2.0f * (float)((b >> (3 - m)) & 1)) : 0.0f;
    }
  }

  const int waves_per_grid = gridDim.x * 8;
  for (int tile = blockIdx.x * 8 + wave; tile < tiles_total; tile += waves_per_grid) {
    const int b_img = tile >> 10;              // 128*8 = 1024 tiles per image
    const int rem   = tile & 1023;
    const int i     = rem >> 3;                // output row
    const int j0    = (rem & 7) << 4;          // first of 16 output cols

    // ---- load this lane's 2x2 window (both halves duplicate window m) ----
    const float* p = x + (((size_t)b_img) << 16) + ((size_t)i << 9)
                       + ((size_t)(j0 + m) << 1);
    v2f p01 = *(const v2f*)(p);
    v2f p23 = *(const v2f*)(p + 256);

    float s0, c0, s1, c1, s2, c2, s3, c3;
    __sincosf(p01[0] * PI_HALF, &s0, &c0);
    __sincosf(p01[1] * PI_HALF, &s1, &c1);
    __sincosf(p23[0] * PI_HALF, &s2, &c2);
    __sincosf(p23[1] * PI_HALF, &s3, &c3);

    // ---- embedded state, real/imag merged with (-i)^popcount sign folded ----
    float F01[4] = {c0 * c1, c0 * s1, s0 * c1, s0 * s1};
    float F23[4] = {c2 * c3, c2 * s3, s2 * c3, s2 * s3};
    float w[16];
#pragma unroll
    for (int b = 0; b < 16; ++b) {
      const int pc = __popc(b);
      const float sg = (pc == 1 || pc == 2) ? -1.0f : 1.0f;
      w[b] = sg * F01[(b >> 2) & 3] * F23[b & 3];
    }

    // ---- B operand: W^T, K chunked by 4 (lane supplies K = 4kk+v+2hi) ----
    v2f Bw[4];
#pragma unroll
    for (int kk = 0; kk < 4; ++kk) {
#pragma unroll
      for (int v = 0; v < 2; ++v)
        Bw[kk][v] = hi ? w[4 * kk + v + 2] : w[4 * kk + v];
    }

    // ---- GEMM1: out_re^T and out_im^T ----
    v8f accRe = {0.f, 0.f, 0.f, 0.f, 0.f, 0.f, 0.f, 0.f};
    v8f accIm = {0.f, 0.f, 0.f, 0.f, 0.f, 0.f, 0.f, 0.f};
#pragma unroll
    for (int kk = 0; kk < 4; ++kk) {
      accRe = __builtin_amdgcn_wmma_f32_16x16x4_f32(false, Ae[kk], false, Bw[kk],
                                                    (short)0, accRe, false, false);
      accIm = __builtin_amdgcn_wmma_f32_16x16x4_f32(false, Ao[kk], false, Bw[kk],
                                                    (short)0, accIm, false, false);
    }

    // ---- probabilities (C/D layout: comp = r + 8*hi, window = lane&15) ----
    v8f P;
#pragma unroll
    for (int r = 0; r < 8; ++r) P[r] = accRe[r] * accRe[r] + accIm[r] * accIm[r];

    // ---- rearrange probs^T from C/D layout into B layout via shuffles ----
    // chunk kk needs K = 4kk+v+2hi -> source half hi' = (kk>=2), VGPR 4*(kk&1)+v+2hi
    v2f Bp[4];
#pragma unroll
    for (int kk = 0; kk < 4; ++kk) {
      const int src = (kk >= 2) ? (m + 16) : m;
      const int rb  = (kk & 1) ? 4 : 0;
#pragma unroll
      for (int v = 0; v < 2; ++v) {
        float plo = __shfl(P[rb + v],     src, 32);
        float phi = __shfl(P[rb + v + 2], src, 32);
        Bp[kk][v] = hi ? phi : plo;
      }
    }

    // ---- GEMM2: EZ^T = Z^T x probs^T ----
    v8f accZ = {0.f, 0.f, 0.f, 0.f, 0.f, 0.f, 0.f, 0.f};
#pragma unroll
    for (int kk = 0; kk < 4; ++kk)
      accZ = __builtin_amdgcn_wmma_f32_16x16x4_f32(false, Az[kk], false, Bp[kk],
                                                   (short)0, accZ, false, false);

    // ---- store: lanes 0-15 hold q=0..3 in VGPRs 0..3, N = window ----
    if (!hi) {
      const size_t base = (((size_t)b_img) << 16) + ((size_t)i << 7) + (size_t)(j0 + m);
#pragma unroll
      for (int q = 0; q < 4; ++q)
        out[base + ((size_t)q << 14)] = accZ[q];
    }
  }
}

extern "C" void kernel_launch(void* const* d_in, const int* in_sizes, int n_in,
                              void* d_out, int out_size, void* d_ws, size_t ws_size,
                              hipStream_t stream) {
  const float* x   = (const float*)d_in[0];   // [B,1,256,256] f32
  const float* wts = (const float*)d_in[1];   // [2,4] f32
  float* out = (float*)d_out;                 // [B,4,128,128] f32
  float* U   = (float*)d_ws;                  // 256 floats scratch

  hipLaunchKernelGGL(build_unitary, dim3(1), dim3(32), 0, stream, wts, U);

  const int Bimg  = in_sizes[0] / 65536;      // 64
  const int tiles = Bimg * 1024;              // B * 128 rows * 8 tiles/row
  const int blocks = (tiles + 7) / 8;         // 8 waves (tiles) per 256-thread block
  hipLaunchKernelGGL(quanv_kernel, dim3(blocks), dim3(256), 0, stream,
                     x, U, out, tiles);
}